// AnisotropicSSIMLoss_2851858285092
// MI455X (gfx1250) — compile-verified
//
#include <hip/hip_runtime.h>

// ---------------------------------------------------------------------------
// Anisotropic 3D SSIM loss, fully fused, WMMA-f32 banded-GEMM blur passes.
// Volumes: (2,1,128,256,256) f32. Separable Gaussian 27(z) x 5(y) x 5(x).
// ---------------------------------------------------------------------------

typedef __attribute__((ext_vector_type(2))) float v2f;
typedef __attribute__((ext_vector_type(8))) float v8f;

#define TZ 16
#define TY 16
#define TX 16
#define HZR 42          // real z halo extent (16 + 26)
#define HZ  44          // padded to multiple of 4 (WMMA K)
#define HY  20          // y halo extent (16 + 4)
#define HX  20          // x halo extent (16 + 4)
#define NP  (HY * HX)   // 400 (y,x) halo columns
#define D_ 128
#define H_ 256
#define W_ 256
#define NVOX (2 * D_ * H_ * W_)   // 16,777,216

// LDS layout (float offsets)
#define OFF_RAW1 0
#define OFF_RAW2 (OFF_RAW1 + HZ * NP)          // +17600
#define OFF_ZOUT (OFF_RAW2 + HZ * NP)          // +17600
#define OFF_YOUT (OFF_ZOUT + TZ * NP + 32)     // +6432 (32-float overrun pad)
#define OFF_RES  (OFF_YOUT + TZ * TY * HX)     // +5120
#define OFF_RED  (OFF_RES + 5 * TZ * TY * TX)  // +20480
#define OFF_BZ   (OFF_RED + 8)                 // 64-entry padded z band taps
#define OFF_B5   (OFF_BZ + 64)                 // 64-entry padded y/x band taps
#define SMEM_FLOATS (OFF_B5 + 64)              // 67368 floats = 269,472 B < 320 KB

__device__ __forceinline__ float fieldVal(int f, float r1, float r2) {
  switch (f) {
    case 0:  return r1;
    case 1:  return r2;
    case 2:  return r1 * r1;
    case 3:  return r2 * r2;
    default: return r1 * r2;
  }
}

__global__ __launch_bounds__(256, 1)
void ssim3d_wmma_kernel(const float* __restrict__ img1,
                        const float* __restrict__ img2,
                        float* __restrict__ accum) {
  extern __shared__ float smem[];
  float* raw1 = smem + OFF_RAW1;   // [HZ][NP]  (zero padded)
  float* raw2 = smem + OFF_RAW2;   // [HZ][NP]
  float* zOut = smem + OFF_ZOUT;   // [TZ][NP] + 32 pad (zeroed)
  float* yOut = smem + OFF_YOUT;   // [TZ][TY][HX]
  float* res  = smem + OFF_RES;    // [5][TZ][TY][TX]
  float* red  = smem + OFF_RED;    // [8]
  float* bz   = smem + OFF_BZ;     // bz[16+d] = wz[d] for d in [0,27), else 0
  float* b5   = smem + OFF_B5;     // b5[16+d] = w5[d] for d in [0,5),  else 0

  const int tid  = threadIdx.x;
  // Wave index is uniform across the wave: pin it to an SGPR so tile loops
  // compile to scalar branches (not EXEC-masked vector loops).
  const int wave = __builtin_amdgcn_readfirstlane(tid >> 5);
  const int lane = tid & 31;
  const int lh   = lane >> 4;      // which 16-lane half
  const int ln   = lane & 15;

  // tile decomposition: 16 x-tiles, 16 y-tiles, 8 z-tiles, 2 batches = 4096
  const int blk = blockIdx.x;
  const int txt = blk & 15;
  const int tyt = (blk >> 4) & 15;
  const int tzt = (blk >> 8) & 7;
  const int bb  = blk >> 11;
  const int z0 = tzt * TZ - 13;
  const int y0 = tyt * TY - 2;
  const int x0 = txt * TX - 2;

  // --- Branch-free padded band tables (sigma = 1.5) -----------------------
  if (tid < 64) {
    float s27 = 0.f, s5 = 0.f;
    for (int i = 0; i < 27; ++i) {
      float d = (float)(i - 13);
      s27 += __expf(-d * d * (1.0f / 4.5f));
    }
    for (int i = 0; i < 5; ++i) {
      float d = (float)(i - 2);
      s5 += __expf(-d * d * (1.0f / 4.5f));
    }
    const int dz = tid - 16;   // d in [-16, 47]
    float vz = 0.f;
    if ((unsigned)dz < 27u) {
      float dd = (float)(dz - 13);
      vz = __expf(-dd * dd * (1.0f / 4.5f)) / s27;
    }
    bz[tid] = vz;
    float v5 = 0.f;
    if ((unsigned)dz < 5u) {
      float dd = (float)(dz - 2);
      v5 = __expf(-dd * dd * (1.0f / 4.5f)) / s5;
    }
    b5[tid] = v5;
  }
  // zero the zOut overrun pad (columns past p=399 read by stage B wide tiles)
  if (tid < 32) zOut[TZ * NP + tid] = 0.f;

  // --- Load raw halos (zero padding outside the volume) -------------------
  const long baseB = (long)bb * ((long)D_ * H_ * W_);
  const bool interior = (z0 >= 0) && (z0 + HZR <= D_) &&
                        (y0 >= 0) && (y0 + HY <= H_) &&
                        (x0 >= 0) && (x0 + HX <= W_);
  if (interior) {
    const long base = baseB + ((long)z0 * H_ + y0) * (long)W_ + x0;
    for (int i = tid; i < HZR * NP; i += 256) {
      const int hz  = i / NP;
      const int rem = i - hz * NP;
      const int hy  = rem / HX;
      const int hx  = rem - hy * HX;
      const long idx = base + ((long)hz * H_ + hy) * (long)W_ + hx;
      raw1[i] = img1[idx];
      raw2[i] = img2[idx];
    }
  } else {
    for (int i = tid; i < HZR * NP; i += 256) {
      const int hz  = i / NP;
      const int rem = i - hz * NP;
      const int hy  = rem / HX;
      const int hx  = rem - hy * HX;
      const int gz = z0 + hz, gy = y0 + hy, gx = x0 + hx;
      float a = 0.f, b = 0.f;
      if (gz >= 0 && gz < D_ && gy >= 0 && gy < H_ && gx >= 0 && gx < W_) {
        const long idx = baseB + ((long)gz * H_ + gy) * (long)W_ + gx;
        a = img1[idx];
        b = img2[idx];
      }
      raw1[i] = a;
      raw2[i] = b;
    }
  }
  // zero the two padded z rows (k = 42, 43)
  for (int i = HZR * NP + tid; i < HZ * NP; i += 256) {
    raw1[i] = 0.f;
    raw2[i] = 0.f;
  }
  __syncthreads();

  // --- Hoisted band fragments (depend only on lane + K-chunk) -------------
  v2f aZ[11];
#pragma unroll
  for (int kc = 0; kc < 11; ++kc) {
    const int ka = kc * 4 + lh * 2;
    aZ[kc].x = bz[16 + ka + 0 - ln];
    aZ[kc].y = bz[16 + ka + 1 - ln];
  }
  v2f a5[5];
#pragma unroll
  for (int kc = 0; kc < 5; ++kc) {
    const int ka = kc * 4 + lh * 2;
    a5[kc].x = b5[16 + ka + 0 - ln];
    a5[kc].y = b5[16 + ka + 1 - ln];
  }

  // --- Per-field separable blur: three banded WMMA GEMM passes ------------
  // Fully unrolled so fieldVal() constant-folds (no scalar switch in loops).
#pragma unroll
  for (int f = 0; f < 5; ++f) {
    // Stage A: Z blur.  zOut[16][400] = BandZ(16x44) * prod(44x400)
    for (int t = wave; t < 25; t += 8) {
      const int p0 = t * 16;
      const int p  = p0 + ln;
      v8f acc = {};
#pragma unroll
      for (int kc = 0; kc < 11; ++kc) {
        const int ka = kc * 4 + lh * 2;
        v2f B;
        const float r10 = raw1[(ka + 0) * NP + p];
        const float r20 = raw2[(ka + 0) * NP + p];
        const float r11 = raw1[(ka + 1) * NP + p];
        const float r21 = raw2[(ka + 1) * NP + p];
        B.x = fieldVal(f, r10, r20);
        B.y = fieldVal(f, r11, r21);
        acc = __builtin_amdgcn_wmma_f32_16x16x4_f32(false, aZ[kc], false, B,
                                                    (short)0, acc, false, false);
      }
#pragma unroll
      for (int j = 0; j < 8; ++j) {
        const int m = j + 8 * lh;
        zOut[m * NP + p0 + ln] = acc[j];
      }
    }
    __syncthreads();

    // Stage B: Y blur.  yOut[z][16][20] = BandY(16x20) * zOut[z](20x20)
    // Columns n >= 20 compute garbage and are discarded by the store guard;
    // their reads land in the zeroed pad past zOut, staying finite.
    for (int job = wave; job < 32; job += 8) {
      const int z  = job >> 1;
      const int xt = job & 1;
      const int n  = xt * 16 + ln;
      v8f acc = {};
#pragma unroll
      for (int kc = 0; kc < 5; ++kc) {
        const int ka = kc * 4 + lh * 2;
        v2f B;
        B.x = zOut[z * NP + (ka + 0) * HX + n];
        B.y = zOut[z * NP + (ka + 1) * HX + n];
        acc = __builtin_amdgcn_wmma_f32_16x16x4_f32(false, a5[kc], false, B,
                                                    (short)0, acc, false, false);
      }
      if (n < HX) {
#pragma unroll
        for (int j = 0; j < 8; ++j) {
          const int m = j + 8 * lh;
          yOut[(z * TY + m) * HX + n] = acc[j];
        }
      }
    }
    __syncthreads();

    // Stage C: X blur.  res rows[(z,y) 16] x16 = yOut rows(16x20) * BandX(20x16)
    for (int rt = wave; rt < 16; rt += 8) {
      const int row = rt * 16 + ln;   // A-matrix row (z*16+y)
      v8f acc = {};
#pragma unroll
      for (int kc = 0; kc < 5; ++kc) {
        const int ka = kc * 4 + lh * 2;
        v2f A;
        A.x = yOut[row * HX + ka + 0];
        A.y = yOut[row * HX + ka + 1];
        acc = __builtin_amdgcn_wmma_f32_16x16x4_f32(false, A, false, a5[kc],
                                                    (short)0, acc, false, false);
      }
#pragma unroll
      for (int j = 0; j < 8; ++j) {
        const int rr = rt * 16 + j + 8 * lh;
        const int z = rr >> 4, y = rr & 15;
        res[((f * TZ + z) * TY + y) * TX + ln] = acc[j];
      }
    }
    __syncthreads();
  }

  // --- SSIM map + block reduction -----------------------------------------
  const float C1c = 0.0001f;    // 0.01^2
  const float C2c = 0.0009f;    // 0.03^2
  float local = 0.f;
  for (int i = tid; i < TZ * TY * TX; i += 256) {
    const float mu1 = res[0 * 4096 + i];
    const float mu2 = res[1 * 4096 + i];
    const float b11 = res[2 * 4096 + i];
    const float b22 = res[3 * 4096 + i];
    const float b12 = res[4 * 4096 + i];
    const float mu1s = mu1 * mu1, mu2s = mu2 * mu2, m12 = mu1 * mu2;
    const float s1 = b11 - mu1s, s2 = b22 - mu2s, s12 = b12 - m12;
    const float num = (2.f * m12 + C1c) * (2.f * s12 + C2c);
    const float den = (mu1s + mu2s + C1c) * (s1 + s2 + C2c);
    local += num / den;
  }
  for (int off = 16; off > 0; off >>= 1)
    local += __shfl_down(local, off, 32);
  if (lane == 0) red[wave] = local;
  __syncthreads();
  if (tid == 0) {
    float s = 0.f;
    for (int wv = 0; wv < 8; ++wv) s += red[wv];
    atomicAdd(accum, s);
  }
}

__global__ void ssim_zero_kernel(float* acc) {
  if (threadIdx.x == 0) acc[0] = 0.f;
}

__global__ void ssim_finalize_kernel(const float* acc, float* out) {
  if (threadIdx.x == 0) out[0] = 1.0f - acc[0] * (1.0f / (float)NVOX);
}

extern "C" void kernel_launch(void* const* d_in, const int* in_sizes, int n_in,
                              void* d_out, int out_size, void* d_ws, size_t ws_size,
                              hipStream_t stream) {
  (void)in_sizes; (void)n_in; (void)out_size; (void)ws_size;
  const float* img1 = (const float*)d_in[0];
  const float* img2 = (const float*)d_in[1];
  float* out = (float*)d_out;
  float* acc = (float*)d_ws;

  ssim_zero_kernel<<<1, 32, 0, stream>>>(acc);
  const size_t smem = (size_t)SMEM_FLOATS * sizeof(float);
  ssim3d_wmma_kernel<<<4096, 256, smem, stream>>>(img1, img2, acc);
  ssim_finalize_kernel<<<1, 32, 0, stream>>>(acc, out);
}